// SparseVariableRouter_46179488367247
// MI455X (gfx1250) — compile-verified
//
#include <hip/hip_runtime.h>
#include <math.h>

#define NUM_VARS 512
#define HIDDEN   16
#define TOPK     8

typedef __attribute__((ext_vector_type(2))) float v2f;
typedef __attribute__((ext_vector_type(8))) float v8f;

// ---------------------------------------------------------------------------
// CDNA5 async memory->LDS path (ASYNCcnt-tracked), with safe fallback.
// The builtin's b128 payload type is a 4 x i32 vector; arg0 = global source
// (AS1, non-const), arg1 = LDS destination (AS3), then imm offset + cpol.
// ---------------------------------------------------------------------------
#if __has_builtin(__builtin_amdgcn_global_load_async_to_lds_b128) && \
    __has_builtin(__builtin_amdgcn_s_wait_asynccnt)
#define HAVE_ASYNC_LDS 1
typedef int v4i __attribute__((ext_vector_type(4)));
typedef __attribute__((address_space(1))) v4i v4i_glb;
typedef __attribute__((address_space(3))) v4i v4i_lds;
__device__ __forceinline__ void async_copy_b128(const float4* g, float4* l) {
    __builtin_amdgcn_global_load_async_to_lds_b128(
        (v4i_glb*)g, (v4i_lds*)l, /*imm offset=*/0, /*cpol=*/0);
}
__device__ __forceinline__ void async_wait_le4() { __builtin_amdgcn_s_wait_asynccnt(4); }
__device__ __forceinline__ void async_wait_0()   { __builtin_amdgcn_s_wait_asynccnt(0); }
#else
#define HAVE_ASYNC_LDS 0
__device__ __forceinline__ void async_copy_b128(const float4* g, float4* l) { *l = *g; }
__device__ __forceinline__ void async_wait_le4() {}
__device__ __forceinline__ void async_wait_0()   {}
#endif

// ---------------------------------------------------------------------------
// Kernel 1: Q = ve @ Wq^T + bq ; K = ve @ Wk^T + bk   (512x16, trivial)
// ---------------------------------------------------------------------------
__global__ void qk_kernel(const float* __restrict__ ve,
                          const float* __restrict__ Wq, const float* __restrict__ bq,
                          const float* __restrict__ Wk, const float* __restrict__ bk,
                          float* __restrict__ Q, float* __restrict__ K) {
    int n = blockIdx.x * blockDim.x + threadIdx.x;
    if (n >= NUM_VARS) return;
    float e[HIDDEN];
#pragma unroll
    for (int j = 0; j < HIDDEN; ++j) e[j] = ve[n * HIDDEN + j];
#pragma unroll
    for (int h = 0; h < HIDDEN; ++h) {
        float q = bq[h];
        float k = bk[h];
#pragma unroll
        for (int j = 0; j < HIDDEN; ++j) {
            q = fmaf(e[j], Wq[h * HIDDEN + j], q);
            k = fmaf(e[j], Wk[h * HIDDEN + j], k);
        }
        Q[n * HIDDEN + h] = q;
        K[n * HIDDEN + h] = k;
    }
}

// ---------------------------------------------------------------------------
// Kernel 2: sim = Q @ K^T  (512x512) via V_WMMA_F32_16X16X4_F32.
// One wave computes one 16x16 tile; K-dim = 16 -> 4 chained WMMAs.
// (Operand layouts per ISA 7.12.2; diagonal masked in the accumulator.)
// ---------------------------------------------------------------------------
__global__ void sim_kernel(const float* __restrict__ Q,
                           const float* __restrict__ K,
                           float* __restrict__ sim) {
    const int wave = (blockIdx.x * blockDim.x + threadIdx.x) >> 5;  // 0..1023
    const int lane = threadIdx.x & 31;
    const int ti = wave >> 5;   // tile row 0..31
    const int tj = wave & 31;   // tile col 0..31
    const int m  = lane & 15;
    const int half = lane >> 4;        // 0 or 1
    const int kk = half * 2;           // K sub-offset within the 4-chunk

    const float* qrow = Q + (ti * 16 + m) * HIDDEN;
    const float* krow = K + (tj * 16 + m) * HIDDEN;

    v8f c = {};
#pragma unroll
    for (int k0 = 0; k0 < HIDDEN; k0 += 4) {
        v2f a, b;
        a.x = qrow[k0 + kk];
        a.y = qrow[k0 + kk + 1];
        b.x = krow[k0 + kk];
        b.y = krow[k0 + kk + 1];
        c = __builtin_amdgcn_wmma_f32_16x16x4_f32(
                /*neg_a=*/false, a, /*neg_b=*/false, b,
                /*c_mod=*/(short)0, c, /*reuse_a=*/false, /*reuse_b=*/false);
    }

#pragma unroll
    for (int r = 0; r < 8; ++r) {
        int row = r + half * 8;
        int col = m;
        float val = c[r];
        if (ti == tj && row == col) val = -1.0e9f;
        sim[(size_t)(ti * 16 + row) * NUM_VARS + (tj * 16 + col)] = val;
    }
}

// ---------------------------------------------------------------------------
// Kernel 3: per-row top-8 (sorted insertion in registers) + softmax.
// ---------------------------------------------------------------------------
__global__ void topk_kernel(const float* __restrict__ sim,
                            float* __restrict__ w_out,
                            int* __restrict__ idx_out) {
    int i = blockIdx.x * blockDim.x + threadIdx.x;
    if (i >= NUM_VARS) return;
    const float* row = sim + (size_t)i * NUM_VARS;

    float v[TOPK];
    int   id[TOPK];
#pragma unroll
    for (int k = 0; k < TOPK; ++k) { v[k] = -3.0e38f; id[k] = 0; }

    for (int j = 0; j < NUM_VARS; ++j) {
        float s = row[j];
        if (s > v[TOPK - 1]) {
            v[TOPK - 1] = s;
            id[TOPK - 1] = j;
#pragma unroll
            for (int k = TOPK - 1; k > 0; --k) {
                if (v[k] > v[k - 1]) {
                    float tv = v[k]; v[k] = v[k - 1]; v[k - 1] = tv;
                    int ti_ = id[k]; id[k] = id[k - 1]; id[k - 1] = ti_;
                }
            }
        }
    }

    float vmax = v[0];
    float e[TOPK];
    float sum = 0.0f;
#pragma unroll
    for (int k = 0; k < TOPK; ++k) { e[k] = __expf(v[k] - vmax); sum += e[k]; }
    float inv = 1.0f / sum;
#pragma unroll
    for (int k = 0; k < TOPK; ++k) {
        w_out[i * TOPK + k]  = e[k] * inv;
        idx_out[i * TOPK + k] = id[k];
    }
}

// ---------------------------------------------------------------------------
// Kernel 4: bandwidth kernel, double-buffered with CDNA5 async LDS loads.
//
// Each block processes NCHUNKS chunks of GROWS contiguous (b,l) rows.
// Chunk c+1 streams memory->LDS via global_load_async_to_lds_b128 while
// chunk c is being gathered. Per wave, one chunk = 4 async instructions
// (in-order completion), so s_wait_asynccnt<=4 retires the current chunk
// while the next chunk's 4 loads stay in flight.
// ---------------------------------------------------------------------------
#define GROWS        8                               // rows per chunk
#define CHUNK_FLOATS (GROWS * NUM_VARS)              // 4096 floats = 16 KB
#define CHUNK_VEC4   (CHUNK_FLOATS / 4)              // 1024 float4
#define NCHUNKS      4                               // chunks per block

__global__ void gather_kernel(const float* __restrict__ x,
                              const float* __restrict__ w,
                              const int* __restrict__ idx,
                              float* __restrict__ out) {
    __shared__ float buf[2][CHUNK_FLOATS];

    const int tid = threadIdx.x;                     // 0..255
    const size_t blockBase = (size_t)blockIdx.x * (CHUNK_FLOATS * NCHUNKS);

    // Per-thread routing tables in registers (n0 = tid, n1 = tid + 256),
    // amortized over all GROWS*NCHUNKS rows this block handles.
    const int n0 = tid, n1 = tid + 256;
    float w0[TOPK], w1[TOPK];
    int   i0[TOPK], i1[TOPK];
#pragma unroll
    for (int k = 0; k < TOPK; ++k) {
        w0[k] = w[n0 * TOPK + k];  i0[k] = idx[n0 * TOPK + k];
        w1[k] = w[n1 * TOPK + k];  i1[k] = idx[n1 * TOPK + k];
    }

    // Issue async loads for one chunk: 4 b128 per thread (4 per-wave instrs).
    auto prefetch = [&](int c, int bsel) {
        const float4* g = (const float4*)(x + blockBase + (size_t)c * CHUNK_FLOATS);
        float4* l = (float4*)buf[bsel];
#pragma unroll
        for (int t = 0; t < CHUNK_VEC4 / 256; ++t) {   // 4 iterations
            int slot = tid + t * 256;
            async_copy_b128(g + slot, l + slot);
        }
    };

    prefetch(0, 0);

    for (int c = 0; c < NCHUNKS; ++c) {
        const int cur = c & 1;
        if (c + 1 < NCHUNKS) {
            prefetch(c + 1, cur ^ 1);
            async_wait_le4();        // current chunk landed; next still in flight
        } else {
            async_wait_0();
        }
        __syncthreads();             // all waves' loads for chunk c visible

        const float* xb = buf[cur];
        float* outc = out + blockBase + (size_t)c * CHUNK_FLOATS;
#pragma unroll
        for (int r = 0; r < GROWS; ++r) {
            const float* xr = xb + r * NUM_VARS;
            float acc0 = 0.0f, acc1 = 0.0f;
#pragma unroll
            for (int k = 0; k < TOPK; ++k) {
                acc0 = fmaf(w0[k], xr[i0[k]], acc0);
                acc1 = fmaf(w1[k], xr[i1[k]], acc1);
            }
            outc[(size_t)r * NUM_VARS + n0] = acc0;
            outc[(size_t)r * NUM_VARS + n1] = acc1;
        }
        __syncthreads();             // protect buf[cur] before it is re-filled
    }
}

// ---------------------------------------------------------------------------
// Launcher
// ---------------------------------------------------------------------------
extern "C" void kernel_launch(void* const* d_in, const int* in_sizes, int n_in,
                              void* d_out, int out_size, void* d_ws, size_t ws_size,
                              hipStream_t stream) {
    const float* x  = (const float*)d_in[0];
    const float* ve = (const float*)d_in[1];
    const float* Wq = (const float*)d_in[2];
    const float* bq = (const float*)d_in[3];
    const float* Wk = (const float*)d_in[4];
    const float* bk = (const float*)d_in[5];
    float* out = (float*)d_out;

    // Workspace layout (bytes):
    //   Q   : [0      , 32768)
    //   K   : [32768  , 65536)
    //   w   : [65536  , 81920)
    //   idx : [81920  , 98304)
    //   sim : [98304  , 98304 + 1 MiB)
    char* ws = (char*)d_ws;
    float* Qw   = (float*)(ws + 0);
    float* Kw   = (float*)(ws + 32768);
    float* ww   = (float*)(ws + 65536);
    int*   idxw = (int*)  (ws + 81920);
    float* simw = (float*)(ws + 98304);

    const int rows = in_sizes[0] / NUM_VARS;   // B*L = 32768

    qk_kernel<<<2, 256, 0, stream>>>(ve, Wq, bq, Wk, bk, Qw, Kw);
    sim_kernel<<<128, 256, 0, stream>>>(Qw, Kw, simw);
    topk_kernel<<<2, 256, 0, stream>>>(simw, ww, idxw);

    const int rows_per_block = GROWS * NCHUNKS;            // 32
    gather_kernel<<<rows / rows_per_block, 256, 0, stream>>>(x, ww, idxw, out);
}